// FFT_Linear_87359634800828
// MI455X (gfx1250) — compile-verified
//
#include <hip/hip_runtime.h>
#include <hip/hip_bf16.h>

// Block-circulant linear via frequency domain, exploiting real-input Hermitian
// symmetry (32 packed frequency slots instead of 64):
//   Xf = FFT64(x blocks), natural order via free compile-time bit-reversal.
//   Slot 0 packs the two purely-real bins: re-plane = Xf[0], im-plane = Xf[32].
//   Slots 1..31 are ordinary complex bins.
//   Of[b,p,slot] = sum_q Xf[b,q,slot]*Wf[p,q,slot] with 4 real WMMA chains
//   (V_WMMA_F32_16X16X4_F32); for slot 0, p_rr = Of[k=0], p_ii = Of[k=32].
//   out = real(IFFT64(Of)) reconstructed from 33 bins per (b,p) sequence.
//
// Sizes: B=4096, Q=P=64 blocks, BS=64. Workspace ~68.5 MB.

#define BATCH 4096
#define NBLK  64
#define BS    64

typedef __attribute__((ext_vector_type(2))) float v2f;
typedef __attribute__((ext_vector_type(8))) float v8f;

// 6-bit bit-reversal (involution); used only at fully-unrolled constant indices.
constexpr int BITREV[64] = {
   0,32,16,48, 8,40,24,56, 4,36,20,52,12,44,28,60,
   2,34,18,50,10,42,26,58, 6,38,22,54,14,46,30,62,
   1,33,17,49, 9,41,25,57, 5,37,21,53,13,45,29,61,
   3,35,19,51,11,43,27,59, 7,39,23,55,15,47,31,63
};

// cos/sin(2*pi*j/64), j = 0..31. Forward twiddle = c - i*s, inverse = c + i*s.
__constant__ float COS64[32] = {
  1.0f, 0.9951847267f, 0.9807852804f, 0.9569403357f,
  0.9238795325f, 0.8819212643f, 0.8314696123f, 0.7730104534f,
  0.7071067812f, 0.6343932842f, 0.5555702330f, 0.4713967368f,
  0.3826834324f, 0.2902846773f, 0.1950903220f, 0.0980171403f,
  0.0f, -0.0980171403f, -0.1950903220f, -0.2902846773f,
  -0.3826834324f, -0.4713967368f, -0.5555702330f, -0.6343932842f,
  -0.7071067812f, -0.7730104534f, -0.8314696123f, -0.8819212643f,
  -0.9238795325f, -0.9569403357f, -0.9807852804f, -0.9951847267f
};
__constant__ float SIN64[32] = {
  0.0f, 0.0980171403f, 0.1950903220f, 0.2902846773f,
  0.3826834324f, 0.4713967368f, 0.5555702330f, 0.6343932842f,
  0.7071067812f, 0.7730104534f, 0.8314696123f, 0.8819212643f,
  0.9238795325f, 0.9569403357f, 0.9807852804f, 0.9951847267f,
  1.0f, 0.9951847267f, 0.9807852804f, 0.9569403357f,
  0.9238795325f, 0.8819212643f, 0.8314696123f, 0.7730104534f,
  0.7071067812f, 0.6343932842f, 0.5555702330f, 0.4713967368f,
  0.3826834324f, 0.2902846773f, 0.1950903220f, 0.0980171403f
};

// Forward radix-2 DIF: natural input, bit-reversed output; register resident.
__device__ __forceinline__ void fft64_dif(float re[64], float im[64]) {
#pragma unroll
  for (int h = 32; h >= 1; h >>= 1) {
    const int tstep = 32 / h;
#pragma unroll
    for (int s = 0; s < 64; s += 2 * h) {
#pragma unroll
      for (int j = 0; j < h; ++j) {
        const float wr = COS64[j * tstep];
        const float wi = -SIN64[j * tstep];
        const float ur = re[s + j],     ui = im[s + j];
        const float vr = re[s + j + h], vi = im[s + j + h];
        re[s + j] = ur + vr;
        im[s + j] = ui + vi;
        const float dr = ur - vr, di = ui - vi;
        re[s + j + h] = dr * wr - di * wi;
        im[s + j + h] = dr * wi + di * wr;
      }
    }
  }
}

// Inverse radix-2 DIT: bit-reversed input, natural output. Caller scales 1/64.
__device__ __forceinline__ void ifft64_dit(float re[64], float im[64]) {
#pragma unroll
  for (int h = 1; h <= 32; h <<= 1) {
    const int tstep = 32 / h;
#pragma unroll
    for (int s = 0; s < 64; s += 2 * h) {
#pragma unroll
      for (int j = 0; j < h; ++j) {
        const float wr = COS64[j * tstep];
        const float wi = SIN64[j * tstep];  // conjugate of forward
        const float vr0 = re[s + j + h], vi0 = im[s + j + h];
        const float vr = vr0 * wr - vi0 * wi;
        const float vi = vr0 * wi + vi0 * wr;
        const float ur = re[s + j], ui = im[s + j];
        re[s + j] = ur + vr;
        im[s + j] = ui + vi;
        re[s + j + h] = ur - vr;
        im[s + j + h] = ui - vi;
      }
    }
  }
}

// Kernel 1: FFT of every 64-wide block of x. 32 packed slot planes, layout
// [slot][q][b] (b innermost -> coalesced stores, coalesced A-fragment loads).
__global__ void __launch_bounds__(256)
fft_blocks_x(const float* __restrict__ x,
             float* __restrict__ XfR, float* __restrict__ XfI) {
  const int b = blockIdx.x * 256 + threadIdx.x;  // 0..4095
  const int q = blockIdx.y;                      // 0..63
  float re[64], im[64];
  const float4* s4 = (const float4*)(x + (size_t)b * (NBLK * BS) + q * BS);
#pragma unroll
  for (int e = 0; e < 16; ++e) {
    const float4 v = s4[e];
    re[4 * e + 0] = v.x; re[4 * e + 1] = v.y;
    re[4 * e + 2] = v.z; re[4 * e + 3] = v.w;
    im[4 * e + 0] = 0.f; im[4 * e + 1] = 0.f;
    im[4 * e + 2] = 0.f; im[4 * e + 3] = 0.f;
  }
  fft64_dif(re, im);
  const size_t PL = (size_t)NBLK * BATCH;        // plane stride
  const size_t base = (size_t)q * BATCH + b;
  // Slot 0: re-plane = freq 0 (DC, at pos BITREV[0]=0), im-plane = freq 32
  // (Nyquist, at pos BITREV[32]=1). Both purely real for real input.
  XfR[base] = re[0];
  XfI[base] = re[1];
#pragma unroll
  for (int j = 1; j < 32; ++j) {
    XfR[j * PL + base] = re[BITREV[j]];
    XfI[j * PL + base] = im[BITREV[j]];
  }
}

// Kernel 2: same for W[p][q][:], layout [slot][q][p] (p innermost -> natural
// B-fragment (Wf^T) loads).
__global__ void __launch_bounds__(256)
fft_blocks_w(const float* __restrict__ W,
             float* __restrict__ WfR, float* __restrict__ WfI) {
  const int t = blockIdx.x * 256 + threadIdx.x;  // 0..4095
  const int p = t & 63;
  const int q = t >> 6;
  float re[64], im[64];
  const float4* s4 = (const float4*)(W + ((size_t)p * NBLK + q) * BS);
#pragma unroll
  for (int e = 0; e < 16; ++e) {
    const float4 v = s4[e];
    re[4 * e + 0] = v.x; re[4 * e + 1] = v.y;
    re[4 * e + 2] = v.z; re[4 * e + 3] = v.w;
    im[4 * e + 0] = 0.f; im[4 * e + 1] = 0.f;
    im[4 * e + 2] = 0.f; im[4 * e + 3] = 0.f;
  }
  fft64_dif(re, im);
  const size_t PL = (size_t)NBLK * NBLK;
  const size_t base = (size_t)q * NBLK + p;
  WfR[base] = re[0];
  WfI[base] = re[1];
#pragma unroll
  for (int j = 1; j < 32; ++j) {
    WfR[j * PL + base] = re[BITREV[j]];
    WfI[j * PL + base] = im[BITREV[j]];
  }
}

// Of-in-LDS layout: padded so both WMMA stash and ifft gather are
// bank-conflict-free. n-stride 33, m-stride 529 (both odd vs 64 banks).
#define OF_NSTRIDE 33
#define OF_MSTRIDE 529                 // 16*33 + 1
#define OF_HALF    (16 * OF_MSTRIDE)   // 8464 floats (one re or im plane)
#define OF_PT      (2 * OF_HALF)       // one p-tile region (re + im)
#define LDS_FLOATS (2 * OF_PT)         // two p-tiles: 33856 floats = 135,424 B

// Kernel 3: per WG: one 16-batch tile x TWO 16-p tiles (A fragments reused
// across both -> Xf read amplification 2x instead of 4x). 8 waves x 4 slots.
__global__ void __launch_bounds__(256)
bcirc_matmul_ifft(const float* __restrict__ XfR, const float* __restrict__ XfI,
                  const float* __restrict__ WfR, const float* __restrict__ WfI,
                  float* __restrict__ out) {
  extern __shared__ float lds[];

  const int b0   = blockIdx.x * 16;        // batch tile
  const int p0   = blockIdx.y * 32;        // base of this WG's two p-tiles
  const int lane = threadIdx.x & 31;
  const int wave = threadIdx.x >> 5;       // 8 waves x 4 slots = 32 slots
  const int lm   = lane & 15;              // A: M row / B: N col
  const int lh   = lane >> 4;              // K sub-pair select (per ISA layout)

  for (int jj = 0; jj < 4; ++jj) {
    const int j = wave * 4 + jj;           // frequency slot 0..31
    const float* xr = XfR + (size_t)j * (NBLK * BATCH);
    const float* xi = XfI + (size_t)j * (NBLK * BATCH);
    const float* wr = WfR + (size_t)j * (NBLK * NBLK);
    const float* wi = WfI + (size_t)j * (NBLK * NBLK);

    v8f accRR[2] = {}, accII[2] = {}, accRI[2] = {}, accIR[2] = {};
#pragma unroll
    for (int s = 0; s < 16; ++s) {         // K = q, 16 steps of 4
      const int q0 = s * 4 + lh * 2;
      v2f ar, ai;
      // A fragment (16x4 f32): lane<16 holds K=q0,q0+1; lane>=16 q0+2,q0+3
      ar.x = xr[(size_t)(q0    ) * BATCH + b0 + lm];
      ar.y = xr[(size_t)(q0 + 1) * BATCH + b0 + lm];
      ai.x = xi[(size_t)(q0    ) * BATCH + b0 + lm];
      ai.y = xi[(size_t)(q0 + 1) * BATCH + b0 + lm];
#pragma unroll
      for (int pt = 0; pt < 2; ++pt) {
        const int pc = p0 + pt * 16 + lm;
        v2f br, bi;                        // B fragment (4x16 f32), Wf^T
        br.x = wr[(size_t)(q0    ) * NBLK + pc];
        br.y = wr[(size_t)(q0 + 1) * NBLK + pc];
        bi.x = wi[(size_t)(q0    ) * NBLK + pc];
        bi.y = wi[(size_t)(q0 + 1) * NBLK + pc];
        accRR[pt] = __builtin_amdgcn_wmma_f32_16x16x4_f32(
            false, ar, false, br, (short)0, accRR[pt], false, false);
        accII[pt] = __builtin_amdgcn_wmma_f32_16x16x4_f32(
            false, ai, false, bi, (short)0, accII[pt], false, false);
        accRI[pt] = __builtin_amdgcn_wmma_f32_16x16x4_f32(
            false, ar, false, bi, (short)0, accRI[pt], false, false);
        accIR[pt] = __builtin_amdgcn_wmma_f32_16x16x4_f32(
            false, ai, false, br, (short)0, accIR[pt], false, false);
      }
    }
    // Stash Of. C layout: VGPR r -> M = r + 8*(lane>=16), N = lane&15.
    // Slot 0 is the packed real pair: p_rr = Of[k=0], p_ii = Of[k=32].
#pragma unroll
    for (int pt = 0; pt < 2; ++pt) {
      float* lR = lds + pt * OF_PT;
      float* lI = lR + OF_HALF;
#pragma unroll
      for (int r = 0; r < 8; ++r) {
        const int m = r + lh * 8;
        const int idx = m * OF_MSTRIDE + lm * OF_NSTRIDE + j;
        if (j == 0) {
          lR[idx] = accRR[pt][r];                  // Of[k=0]  (real)
          lI[idx] = accII[pt][r];                  // Of[k=32] (real)
        } else {
          lR[idx] = accRR[pt][r] - accII[pt][r];   // Re Of[k=j]
          lI[idx] = accRI[pt][r] + accIR[pt][r];   // Im Of[k=j]
        }
      }
    }
  }
  __syncthreads();

  // Inverse FFT: one (b,p) sequence per thread per p-tile. Reconstruct all 64
  // bins from the 33 stored ones (Hermitian mirror), permute to bit-reversed
  // positions (free, constant indices), DIT back to time domain.
  const int bb = threadIdx.x >> 4;
  const int pp = threadIdx.x & 15;
#pragma unroll 1
  for (int pt = 0; pt < 2; ++pt) {
    const float* lR = lds + pt * OF_PT;
    const float* lI = lR + OF_HALF;
    const int base = bb * OF_MSTRIDE + pp * OF_NSTRIDE;
    float sre[32], sim[32];
#pragma unroll
    for (int u = 0; u < 32; ++u) { sre[u] = lR[base + u]; sim[u] = lI[base + u]; }
    float re[64], im[64];
#pragma unroll
    for (int pos = 0; pos < 64; ++pos) {
      const int k = BITREV[pos];           // DIT input: pos holds freq bitrev(pos)
      float fr, fi;
      if (k == 0)       { fr = sre[0];      fi = 0.0f; }
      else if (k == 32) { fr = sim[0];      fi = 0.0f; }
      else if (k < 32)  { fr = sre[k];      fi = sim[k]; }
      else              { fr = sre[64 - k]; fi = -sim[64 - k]; }
      re[pos] = fr; im[pos] = fi;
    }
    ifft64_dit(re, im);
    float* dst = out + (size_t)(b0 + bb) * (NBLK * BS)
                     + (size_t)(p0 + pt * 16 + pp) * BS;
#pragma unroll
    for (int t = 0; t < 16; ++t) {
      float4 v;
      v.x = re[4 * t + 0] * (1.0f / 64.0f);
      v.y = re[4 * t + 1] * (1.0f / 64.0f);
      v.z = re[4 * t + 2] * (1.0f / 64.0f);
      v.w = re[4 * t + 3] * (1.0f / 64.0f);
      ((float4*)dst)[t] = v;
    }
  }
}

extern "C" void kernel_launch(void* const* d_in, const int* in_sizes, int n_in,
                              void* d_out, int out_size, void* d_ws, size_t ws_size,
                              hipStream_t stream) {
  (void)in_sizes; (void)n_in; (void)out_size; (void)ws_size;
  const float* x = (const float*)d_in[0];   // [4096, 4096]
  const float* W = (const float*)d_in[1];   // [64, 64, 64]
  float* out = (float*)d_out;               // [4096, 4096]

  // Workspace carve-up (~68.5 MB total).
  float* XfR = (float*)d_ws;
  float* XfI = XfR + (size_t)32 * NBLK * BATCH;   // 32 slots * 64q * 4096b
  float* WfR = XfI + (size_t)32 * NBLK * BATCH;
  float* WfI = WfR + (size_t)32 * NBLK * NBLK;    // 32 slots * 64q * 64p

  fft_blocks_x<<<dim3(BATCH / 256, NBLK), 256, 0, stream>>>(x, XfR, XfI);
  fft_blocks_w<<<dim3((NBLK * NBLK) / 256, 1), 256, 0, stream>>>(W, WfR, WfI);

  const size_t shbytes = (size_t)LDS_FLOATS * sizeof(float);  // 135,424 B
  bcirc_matmul_ifft<<<dim3(BATCH / 16, NBLK / 32), 256, shbytes, stream>>>(
      XfR, XfI, WfR, WfI, out);
}